// Word2VectorModelHierarchicalSoftmax_84756884620030
// MI455X (gfx1250) — compile-verified
//
#include <hip/hip_runtime.h>
#include <hip/hip_bf16.h>
#include <math.h>

typedef __attribute__((ext_vector_type(2))) float v2f;
typedef __attribute__((ext_vector_type(8))) float v8f;

// Pointee type must match the builtin's parameter exactly:
//   "__attribute__((__vector_size__(4 * sizeof(int)))) int *"
typedef int v4i __attribute__((vector_size(4 * sizeof(int))));
typedef __attribute__((address_space(1))) v4i* gv4_t;   // global src
typedef __attribute__((address_space(3))) v4i* sv4_t;   // LDS dst

#define B_  2048
#define V_  50000
#define D_  256
#define L_  32
#define BK  40   // 50000 = 40 * 1250; multiple of WMMA K=4
#define PP  44   // padded LDS pitch (floats): 44*m mod 64 distinct for m<16; row=176B (16B-mult)
#define KT  (V_ / BK)   // 1250

// ---------------------------------------------------------------------------
// Kernel 1: X[B,D] = inputs[B,V] @ W[D,V]^T  via V_WMMA_F32_16X16X4_F32
// grid (B/16, D/128), block 256 = 8 waves; wave w owns the 16x16 tile at
// N = nb + 16w.  Staging: GLOBAL_LOAD_ASYNC_TO_LDS_B128 (ASYNCcnt) into
// double-buffered LDS; one s_wait_asynccnt + one barrier per K-tile.
// ---------------------------------------------------------------------------
__global__ __launch_bounds__(256)
void hs_gemm_xw(const float* __restrict__ A,   // [B, V]
                const float* __restrict__ W,   // [D, V]
                float* __restrict__ X)         // [B, D]
{
    __shared__ float As[2][16 * PP];
    __shared__ float Bs[2][128 * PP];

    const int tid  = threadIdx.x;
    const int lane = tid & 31;
    const int wv   = tid >> 5;
    const int m0   = blockIdx.x * 16;
    const int nb   = blockIdx.y * 128;

    // --- async staging maps (all float4 / b128 granularity, 16B aligned) ---
    // A tile: 16 rows x 40 cols = 160 float4.  Lanes 0..19 of every wave issue
    // exactly one b128 -> every wave has identical ASYNCcnt traffic.
    const int afi = wv * 20 + lane;          // valid when lane < 20
    const int ar  = afi / 10, ac4 = afi - 10 * ar;      // row 0..15, f4-col 0..9
    // W tile: 128 rows x 40 cols = 1280 float4, 5 per thread.
    // fi = tid + p*256 ; r = fi/10, c4 = fi%10

    const float* Abase = A + (size_t)(m0 + ar) * V_ + ac4 * 4;

    auto issue_tile = [&](int k0, int buf) {
        // A tile
        if (lane < 20) {
            __builtin_amdgcn_global_load_async_to_lds_b128(
                (gv4_t)(Abase + k0),
                (sv4_t)&As[buf][ar * PP + ac4 * 4], 0, 0);
        }
        // W tile
        #pragma unroll
        for (int p = 0; p < 5; ++p) {
            const int fi = tid + p * 256;
            const int r  = fi / 10;
            const int c4 = fi - 10 * r;
            __builtin_amdgcn_global_load_async_to_lds_b128(
                (gv4_t)(W + (size_t)(nb + r) * V_ + k0 + c4 * 4),
                (sv4_t)&Bs[buf][r * PP + c4 * 4], 0, 0);
        }
    };

    v8f c = {0.f, 0.f, 0.f, 0.f, 0.f, 0.f, 0.f, 0.f};

    // Fragment addressing for V_WMMA_F32_16X16X4_F32:
    // A (16x4): lanes 0-15 hold K={0,1}, lanes 16-31 hold K={2,3}
    const int mrow = lane & 15;
    const int koff = (lane >> 4) * 2;
    const int nrow = wv * 16 + (lane & 15);   // B (4x16) mirrored layout

    issue_tile(0, 0);                          // prologue: tile 0 in flight

    for (int it = 0; it < KT; ++it) {
        const int cur = it & 1;
        asm volatile("s_wait_asynccnt 0x0" ::: "memory");  // tile `it` resident
        __syncthreads();   // everyone done reading buffer cur^1 -> safe to refill
        if (it + 1 < KT) issue_tile((it + 1) * BK, cur ^ 1);

        #pragma unroll
        for (int kk = 0; kk < BK; kk += 4) {
            v2f a, b;
            a.x = As[cur][mrow * PP + kk + koff];
            a.y = As[cur][mrow * PP + kk + koff + 1];
            b.x = Bs[cur][nrow * PP + kk + koff];
            b.y = Bs[cur][nrow * PP + kk + koff + 1];
            c = __builtin_amdgcn_wmma_f32_16x16x4_f32(false, a, false, b,
                                                      (short)0, c, false, false);
        }
    }

    // C/D layout: VGPR r -> M=r (lanes 0-15), M=r+8 (lanes 16-31); N = lane&15
    const int n     = nb + wv * 16 + (lane & 15);
    const int rbase = (lane >> 4) * 8;
    #pragma unroll
    for (int r = 0; r < 8; ++r) {
        X[(size_t)(m0 + rbase + r) * D_ + n] = c[r];
    }
}

// ---------------------------------------------------------------------------
// Kernel 2: per-sample logits + masked BCE mean.  One block (8 waves) per b;
// wave w computes paths j = w, w+8, w+16, w+24 (coalesced 256-dim dots).
// ---------------------------------------------------------------------------
__global__ __launch_bounds__(256)
void hs_path_bce(const float* __restrict__ X,      // [B, D]
                 const float* __restrict__ cls_w,  // [N, D]
                 const int* __restrict__ nodes,    // [B, L]
                 const int* __restrict__ codes,    // [B, L]
                 const int* __restrict__ lens,     // [B]
                 float* __restrict__ per_sample)   // [B]
{
    __shared__ float wsum[8];
    const int b    = blockIdx.x;
    const int lane = threadIdx.x & 31;
    const int wv   = threadIdx.x >> 5;

    const float* xb = X + (size_t)b * D_;
    const float4 x0 = *(const float4*)&xb[lane * 8];
    const float4 x1 = *(const float4*)&xb[lane * 8 + 4];

    const int l_raw = lens[b];
    const int len   = (l_raw > 1) ? l_raw : 1;

    float acc = 0.f;
    #pragma unroll
    for (int p = 0; p < 4; ++p) {
        const int j    = wv + p * 8;
        const int node = nodes[b * L_ + j];
        const float* cw = cls_w + (size_t)node * D_;
        const float4 c0 = *(const float4*)&cw[lane * 8];
        const float4 c1 = *(const float4*)&cw[lane * 8 + 4];
        float s = x0.x*c0.x + x0.y*c0.y + x0.z*c0.z + x0.w*c0.w
                + x1.x*c1.x + x1.y*c1.y + x1.z*c1.z + x1.w*c1.w;
        #pragma unroll
        for (int off = 16; off > 0; off >>= 1)
            s += __shfl_xor(s, off, 32);
        const float z   = s;
        const float y   = (float)codes[b * L_ + j];
        const float bce = fmaxf(z, 0.f) - z * y + log1pf(expf(-fabsf(z)));
        acc += (j < len) ? bce : 0.f;
    }
    if (lane == 0) wsum[wv] = acc;
    __syncthreads();
    if (threadIdx.x == 0) {
        float t = 0.f;
        #pragma unroll
        for (int i = 0; i < 8; ++i) t += wsum[i];
        per_sample[b] = t / (float)len;
    }
}

// ---------------------------------------------------------------------------
// Kernel 3: deterministic single-block mean over B samples -> scalar loss
// ---------------------------------------------------------------------------
__global__ __launch_bounds__(256)
void hs_mean(const float* __restrict__ per_sample, float* __restrict__ out)
{
    __shared__ float red[256];
    float s = 0.f;
    for (int i = threadIdx.x; i < B_; i += 256) s += per_sample[i];
    red[threadIdx.x] = s;
    __syncthreads();
    for (int stride = 128; stride > 0; stride >>= 1) {
        if ((int)threadIdx.x < stride) red[threadIdx.x] += red[threadIdx.x + stride];
        __syncthreads();
    }
    if (threadIdx.x == 0) out[0] = red[0] / (float)B_;
}

extern "C" void kernel_launch(void* const* d_in, const int* in_sizes, int n_in,
                              void* d_out, int out_size, void* d_ws, size_t ws_size,
                              hipStream_t stream) {
    const float* inputs = (const float*)d_in[0];   // [B, V]
    const float* W      = (const float*)d_in[1];   // [D, V]
    const float* cls_w  = (const float*)d_in[2];   // [N, D]
    const int*   nodes  = (const int*)d_in[3];     // [B, L]
    const int*   codes  = (const int*)d_in[4];     // [B, L]
    const int*   lens   = (const int*)d_in[5];     // [B]
    float* out = (float*)d_out;

    float* X          = (float*)d_ws;               // B*D floats (2 MB)
    float* per_sample = X + (size_t)B_ * D_;        // B floats

    dim3 g1(B_ / 16, D_ / 128);                     // 128 x 2 = 256 workgroups
    hs_gemm_xw<<<g1, 256, 0, stream>>>(inputs, W, X);
    hs_path_bce<<<B_, 256, 0, stream>>>(X, cls_w, nodes, codes, lens, per_sample);
    hs_mean<<<1, 256, 0, stream>>>(per_sample, out);
}